// GroupedQueryAttention_56959856279850
// MI455X (gfx1250) — compile-verified
//
#include <hip/hip_runtime.h>
#include <math.h>

// ---------------------------------------------------------------------------
// Grouped Query Attention for MI455X (gfx1250), exact-fp32 WMMA pipeline with
// CDNA5 async-to-LDS staging (GLOBAL_LOAD_ASYNC_TO_LDS_B128/B32 + s_wait_asynccnt).
//   V_WMMA_F32_16X16X4_F32:  A 16x4 (2 VGPR), B 4x16 (2 VGPR), C/D 16x16 (8 VGPR)
// Layouts per CDNA5 ISA:
//   A: lane l holds M = l&15, K = 2*(l>>4) + vgpr        (vgpr = float2 comp)
//   B: lane l holds N = l&15, K = vgpr + 2*(l>>4)
//   C: vgpr r, lane l holds M = r + 8*(l>>4), N = l&15
// K-major operands are stored transposed in LDS (n-major, k rows padded 16->20)
// so every fragment is one aligned, bank-conflict-free ds_load_b64. Fragment
// loads are software-pipelined one k-step ahead of their consuming WMMAs.
// ---------------------------------------------------------------------------

typedef __attribute__((ext_vector_type(2))) float v2f;
typedef __attribute__((ext_vector_type(8))) float v8f;

#define WMMA_F32(a, b, c) \
  __builtin_amdgcn_wmma_f32_16x16x4_f32(false, (a), false, (b), (short)0, (c), false, false)

constexpr int kDModel = 1024;
constexpr int kHeads  = 16;
constexpr int kGroups = 4;
constexpr int kHpg    = 4;   // heads per group
constexpr int kDk     = 64;
constexpr int kB      = 2;
constexpr int kS      = 2048;

// Async copies global -> LDS, tracked by ASYNCcnt (CDNA5). GV addressing mode:
// 64-bit global address in a VGPR pair, LDS byte offset in a VGPR.
__device__ __forceinline__ void async_copy_f4(const float* gsrc, void* ldsDst) {
  const unsigned lds = (unsigned)(uintptr_t)ldsDst;           // addr[31:0] = LDS offset
  const unsigned long long ga = (unsigned long long)(uintptr_t)gsrc;
  asm volatile("global_load_async_to_lds_b128 %0, %1, off"
               :: "v"(lds), "v"(ga) : "memory");
}
__device__ __forceinline__ void async_copy_f1(const float* gsrc, void* ldsDst) {
  const unsigned lds = (unsigned)(uintptr_t)ldsDst;
  const unsigned long long ga = (unsigned long long)(uintptr_t)gsrc;
  asm volatile("global_load_async_to_lds_b32 %0, %1, off"
               :: "v"(lds), "v"(ga) : "memory");
}
__device__ __forceinline__ void wait_async_all() {
  asm volatile("s_wait_asynccnt 0x0" ::: "memory");
}

// ---------------------------------------------------------------------------
// C = A*B + bias GEMM. Block = 256 threads (8 waves), tile 128x64, K-chunk 16.
// Double-buffered LDS tiles filled with async loads; one barrier per chunk.
// B tile is stored transposed via async b32 scatter, so B fragments are b64.
// grid.z indexes (batch, group): z = b*G + g. All shapes divide tiles exactly.
// ---------------------------------------------------------------------------
__global__ __launch_bounds__(256) void gemm_bias_wmma(
    const float* __restrict__ A, int lda, long aBatchStride,
    const float* __restrict__ Bm, int ldb, long bGroupStride,
    const float* __restrict__ bias, int biasGroupStride,
    float* __restrict__ C, int ldc, long cZStride,
    int K, int G) {
  const int z = blockIdx.z;
  A    += (long)(z / G) * aBatchStride;
  Bm   += (long)(z % G) * bGroupStride;
  bias += (long)(z % G) * biasGroupStride;
  C    += (long)z * cZStride;

  const int tid  = threadIdx.x;
  const int wave = tid >> 5;
  const int lane = tid & 31;
  const int half = lane >> 4;   // 0: lanes 0-15, 1: lanes 16-31
  const int lq   = lane & 15;

  const int rowBlock = blockIdx.x * 128;
  const int colBase  = blockIdx.y * 64;

  // aT: A rows x k, row-major [128][16+4]  (80B rows; 20*lq mod 64 distinct)
  // bT: B transposed, n-major [64][16+4]   (fragments: b64 at n*20 + ka)
  __shared__ __attribute__((aligned(16))) float aT[2][128][20];
  __shared__ __attribute__((aligned(16))) float bT[2][64][20];

  // Cooperative load assignment per 16-wide K chunk:
  //   A: 512 float4; thread t -> rows t>>2 and (t>>2)+64, col quarter t&3.
  //   B: thread t -> global row kr = t>>4, n = (t&15)*4 + i, i = 0..3 (b32 scatter).
  const int aRow = tid >> 2;
  const int aCol = (tid & 3) * 4;
  const int bKr  = tid >> 4;
  const int bN   = (tid & 15) * 4;

  const float* aG = A + (long)(rowBlock + aRow) * lda + aCol;
  const float* bG = Bm + (long)bKr * ldb + colBase + bN;
  const long aG2 = (long)64 * lda;  // second A row handled by this thread

  const float* aRowL = &aT[0][wave * 16 + lq][0];   // per-buffer via cur offset
  v8f acc[4] = {};

  // Prologue: stage chunk 0.
  async_copy_f4(aG, &aT[0][aRow][aCol]);
  async_copy_f4(aG + aG2, &aT[0][aRow + 64][aCol]);
#pragma unroll
  for (int i = 0; i < 4; ++i) async_copy_f1(bG + i, &bT[0][bN + i][bKr]);
  wait_async_all();
  __syncthreads();

  for (int k0 = 0; k0 < K; k0 += 16) {
    const int cur = (k0 >> 4) & 1;
    if (k0 + 16 < K) {  // stage next chunk into the other buffer
      async_copy_f4(aG + (k0 + 16), &aT[cur ^ 1][aRow][aCol]);
      async_copy_f4(aG + aG2 + (k0 + 16), &aT[cur ^ 1][aRow + 64][aCol]);
      const float* bn = bG + (long)(k0 + 16) * ldb;
#pragma unroll
      for (int i = 0; i < 4; ++i) async_copy_f1(bn + i, &bT[cur ^ 1][bN + i][bKr]);
    }

    // Software-pipelined fragment loads: stage s+1 loads issued before the
    // stage-s WMMAs, so ds->wmma dependency distance is ~5 loads.
    v2f aF[2], bF[2][4];
    aF[0] = *(const v2f*)(&aT[cur][wave * 16 + lq][2 * half]);
#pragma unroll
    for (int c = 0; c < 4; ++c)
      bF[0][c] = *(const v2f*)(&bT[cur][c * 16 + lq][2 * half]);
#pragma unroll
    for (int s = 0; s < 4; ++s) {
      const int pc = s & 1, pn = pc ^ 1;
      if (s < 3) {
        const int ka = (s + 1) * 4 + 2 * half;
        aF[pn] = *(const v2f*)(&aT[cur][wave * 16 + lq][ka]);
#pragma unroll
        for (int c = 0; c < 4; ++c)
          bF[pn][c] = *(const v2f*)(&bT[cur][c * 16 + lq][ka]);
      }
#pragma unroll
      for (int c = 0; c < 4; ++c) acc[c] = WMMA_F32(aF[pc], bF[pc][c], acc[c]);
    }

    wait_async_all();
    __syncthreads();
  }
  (void)aRowL;

#pragma unroll
  for (int c = 0; c < 4; ++c) {
    const int col = colBase + c * 16 + lq;
    const float bv = bias[col];
#pragma unroll
    for (int r = 0; r < 8; ++r) {
      const int row = rowBlock + wave * 16 + r + 8 * half;
      C[(long)row * ldc + col] = acc[c][r] + bv;
    }
  }
}

// ---------------------------------------------------------------------------
// Flash attention: one block per (128-query tile, head, batch); 8 waves, each
// owning 16 query rows with the 16x64 Q fragment register-resident. K tile is
// row-major in LDS (fragments contiguous b64); V tile is stored transposed so
// its fragments are b64 too. Online softmax with cross-lane reductions.
// ---------------------------------------------------------------------------
__global__ __launch_bounds__(256) void gqa_flash_wmma(
    const float* __restrict__ Q,    // [B, S, 1024] (heads interleaved)
    const float* __restrict__ Kt,   // [B, G, S, 64]
    const float* __restrict__ Vt,   // [B, G, S, 64]
    float* __restrict__ ctx) {      // [B, S, 1024]
  const int b = blockIdx.z;
  const int h = blockIdx.y;
  const int g = h / kHpg;

  const int tid  = threadIdx.x;
  const int wave = tid >> 5;
  const int lane = tid & 31;
  const int half = lane >> 4;
  const int lq   = lane & 15;
  const int qBase = blockIdx.x * 128 + wave * 16;

  const float* Qh = Q + ((long)b * kS) * kDModel + h * kDk;
  const float* Kg = Kt + ((long)(b * kGroups + g)) * kS * kDk;
  const float* Vg = Vt + ((long)(b * kGroups + g)) * kS * kDk;

  // Q tile as 16 A-fragments: qf[c] covers d = 4c + 2*half + {0,1}
  v2f qf[16];
  {
    const float* qrow = Qh + (long)(qBase + lq) * kDModel;
#pragma unroll
    for (int c = 0; c < 16; ++c) qf[c] = *(const v2f*)(qrow + 4 * c + 2 * half);
  }

  // kBuf: K rows x d, row-major [16][64+4]  (fragments b64 at lq*68 + d)
  // vT:   V transposed, dk-major [64][16+4] (fragments b64 at dk*20 + kv)
  __shared__ __attribute__((aligned(16))) float kBuf[2][16][68];
  __shared__ __attribute__((aligned(16))) float vT[2][64][20];
  // Per-wave 16x18 probability tile (C-layout -> A-layout; b64 fragments).
  __shared__ __attribute__((aligned(16))) float pLds[8][16][18];
  float (*pw)[18] = pLds[wave];

  // Cooperative tile loads: thread t -> row t>>4, quarter (t&15)*4.
  const int crow = tid >> 4;
  const int ccol = (tid & 15) * 4;

  v8f acc[4] = {};
  float mrun[8], lrun[8];
#pragma unroll
  for (int r = 0; r < 8; ++r) { mrun[r] = -INFINITY; lrun[r] = 0.f; }

  const float scale = 0.125f;  // 1/sqrt(64)

  // Prologue: stage tile 0 (K row-major b128; V transposed b32 scatter).
  async_copy_f4(Kg + (long)crow * kDk + ccol, &kBuf[0][crow][ccol]);
#pragma unroll
  for (int i = 0; i < 4; ++i)
    async_copy_f1(Vg + (long)crow * kDk + ccol + i, &vT[0][ccol + i][crow]);
  wait_async_all();
  __syncthreads();

  for (int kv0 = 0; kv0 < kS; kv0 += 16) {
    const int cur = (kv0 >> 4) & 1;
    if (kv0 + 16 < kS) {
      const long gofs = (long)(kv0 + 16 + crow) * kDk + ccol;
      async_copy_f4(Kg + gofs, &kBuf[cur ^ 1][crow][ccol]);
#pragma unroll
      for (int i = 0; i < 4; ++i)
        async_copy_f1(Vg + gofs + i, &vT[cur ^ 1][ccol + i][crow]);
    }

    // ---- scores S = Q @ K^T (C layout: M=q, N=kv) -------------------------
    // B fragment: B[d][kv=lq] = K[lq][d], d = 4c + 2*half + {0,1} -> LDS b64.
    // One-fragment lookahead; the 16 WMMAs form one accumulate chain.
    v8f s = {};
    {
      v2f kF[2];
      kF[0] = *(const v2f*)(&kBuf[cur][lq][2 * half]);
#pragma unroll
      for (int c = 0; c < 16; ++c) {
        const int pc = c & 1, pn = pc ^ 1;
        if (c < 15)
          kF[pn] = *(const v2f*)(&kBuf[cur][lq][4 * (c + 1) + 2 * half]);
        s = WMMA_F32(qf[c], kF[pc], s);
      }
    }

    // ---- online softmax over kv (row = 16 lanes of this half-wave) --------
#pragma unroll
    for (int r = 0; r < 8; ++r) {
      float sv = s[r] * scale;
      float m = sv;
      m = fmaxf(m, __shfl_xor(m, 1, 32));
      m = fmaxf(m, __shfl_xor(m, 2, 32));
      m = fmaxf(m, __shfl_xor(m, 4, 32));
      m = fmaxf(m, __shfl_xor(m, 8, 32));
      const float mnew = fmaxf(mrun[r], m);
      const float corr = __expf(mrun[r] - mnew);   // 0 on first tile
      const float p = __expf(sv - mnew);
      float ls = p;
      ls += __shfl_xor(ls, 1, 32);
      ls += __shfl_xor(ls, 2, 32);
      ls += __shfl_xor(ls, 4, 32);
      ls += __shfl_xor(ls, 8, 32);
      lrun[r] = lrun[r] * corr + ls;
      mrun[r] = mnew;
      acc[0][r] *= corr;
      acc[1][r] *= corr;
      acc[2][r] *= corr;
      acc[3][r] *= corr;
      pw[r + 8 * half][lq] = p;  // P[q][kv], q = r + 8*half
    }
    // Same-wave LDS RAW on pLds: DS ops are in-order per wave; the compiler
    // inserts s_wait_dscnt. No cross-wave sharing of pLds -> no barrier here.

    // ---- ctx += P @ V (software-pipelined fragments) ----------------------
    {
      v2f pF[2], vF[2][4];
      pF[0] = *(const v2f*)(&pw[lq][2 * half]);
#pragma unroll
      for (int c = 0; c < 4; ++c)
        vF[0][c] = *(const v2f*)(&vT[cur][c * 16 + lq][2 * half]);
#pragma unroll
      for (int kc = 0; kc < 4; ++kc) {
        const int pc = kc & 1, pn = pc ^ 1;
        if (kc < 3) {
          const int kvr = (kc + 1) * 4 + 2 * half;
          pF[pn] = *(const v2f*)(&pw[lq][kvr]);
#pragma unroll
          for (int c = 0; c < 4; ++c)
            vF[pn][c] = *(const v2f*)(&vT[cur][c * 16 + lq][kvr]);
        }
#pragma unroll
        for (int c = 0; c < 4; ++c) acc[c] = WMMA_F32(pF[pc], vF[pc][c], acc[c]);
      }
    }

    wait_async_all();
    __syncthreads();
  }

  // ---- normalize + store ctx[b, q, h*64 + dk] -----------------------------
  float* crow2 = ctx + ((long)b * kS) * kDModel + h * kDk;
#pragma unroll
  for (int c = 0; c < 4; ++c) {
#pragma unroll
    for (int r = 0; r < 8; ++r) {
      const int q = qBase + r + 8 * half;
      crow2[(long)q * kDModel + c * 16 + lq] = acc[c][r] / lrun[r];
    }
  }
}

// ---------------------------------------------------------------------------
extern "C" void kernel_launch(void* const* d_in, const int* in_sizes, int n_in,
                              void* d_out, int out_size, void* d_ws, size_t ws_size,
                              hipStream_t stream) {
  (void)in_sizes; (void)n_in; (void)out_size; (void)ws_size;

  const float* x   = (const float*)d_in[0];
  const float* W_Q = (const float*)d_in[1];
  const float* b_Q = (const float*)d_in[2];
  const float* W_K = (const float*)d_in[3];
  const float* b_K = (const float*)d_in[4];
  const float* W_V = (const float*)d_in[5];
  const float* b_V = (const float*)d_in[6];
  const float* W_O = (const float*)d_in[7];
  const float* b_O = (const float*)d_in[8];
  float* out = (float*)d_out;

  // Workspace (floats): Q[B,S,1024] | K[B,G,S,64] | V[B,G,S,64] | ctx[B,S,1024]
  float* Qws = (float*)d_ws;
  float* Kws = Qws + (size_t)kB * kS * kDModel;
  float* Vws = Kws + (size_t)kB * kGroups * kS * kDk;
  float* Cws = Vws + (size_t)kB * kGroups * kS * kDk;

  const dim3 blk(256);

  // Q = x @ W_Q + b_Q : M = 4096, N = 1024, K = 1024
  gemm_bias_wmma<<<dim3((kB * kS) / 128, kDModel / 64, 1), blk, 0, stream>>>(
      x, kDModel, 0,
      W_Q, kDModel, 0,
      b_Q, 0,
      Qws, kDModel, 0,
      kDModel, 1);

  // K[b,g] = x[b] @ W_K[g] + b_K[g] : per z=(b*G+g), M = 2048, N = 64, K = 1024
  gemm_bias_wmma<<<dim3(kS / 128, 1, kB * kGroups), blk, 0, stream>>>(
      x, kDModel, (long)kS * kDModel,
      W_K, kDk, (long)kDModel * kDk,
      b_K, kDk,
      Kws, kDk, (long)kS * kDk,
      kDModel, kGroups);

  // V[b,g] = x[b] @ W_V[g] + b_V[g]
  gemm_bias_wmma<<<dim3(kS / 128, 1, kB * kGroups), blk, 0, stream>>>(
      x, kDModel, (long)kS * kDModel,
      W_V, kDk, (long)kDModel * kDk,
      b_V, kDk,
      Vws, kDk, (long)kS * kDk,
      kDModel, kGroups);

  // ctx = softmax(Q K^T / 8) V   (flash, per head)
  gqa_flash_wmma<<<dim3(kS / 128, kHeads, kB), blk, 0, stream>>>(Qws, Kws, Vws, Cws);

  // out = ctx @ W_O + b_O : M = 4096, N = 1024, K = 1024
  gemm_bias_wmma<<<dim3((kB * kS) / 128, kDModel / 64, 1), blk, 0, stream>>>(
      Cws, kDModel, 0,
      W_O, kDModel, 0,
      b_O, 0,
      out, kDModel, 0,
      kDModel, 1);
}